// FFTLongConv1D_60181081751727
// MI455X (gfx1250) — compile-verified
//
#include <hip/hip_runtime.h>

// ---------------------------------------------------------------------------
// FFT long convolution via four-step FFT (N = 16384 = 128 x 128) where every
// DFT-128 stage is a 128x128x128 complex GEMM executed with
// V_WMMA_F32_16X16X4_F32 (full fp32 precision matrix pipe, CDNA5/gfx1250).
//
// Reference semantics replicated exactly, including row_filter: output row
// (b,d) is convolved with h[(b*256 + d) >> 3].
//
// Workspace layout (floats):
//   [0*N)        Fr   : Re DFT128 matrix, F[r][c] = cos(2*pi*r*c/128)
//   [1*N)        Fi   : Im DFT128 matrix,          = -sin(...)
//   [2*N)        Tr   : Re fwd twiddle  W_N^{r*c}
//   [3*N)        Ti   : Im fwd twiddle
//   [4*N)        Tcr  : Re conj twiddle / N   (inverse + scaling folded)
//   [5*N)        Tci  : Im conj twiddle / N
//   [6*N ...)    H spectra: per filter fd: Hr[N], Hi[N]  (256 * 2 * N floats)
// Total = (6 + 512) * 16384 floats ~= 34 MB  (assumed <= ws_size; stays hot
// in the 192 MB L2).
// ---------------------------------------------------------------------------

typedef float v2f __attribute__((ext_vector_type(2)));
typedef float v8f __attribute__((ext_vector_type(8)));

#define NROW   128
#define NFFT   16384      // NROW * NROW
#define PITCH  130        // LDS row pitch (pad to dodge bank conflicts)
#define PLANE  (NROW * PITCH)
#define WGSZ   512        // 16 wave32
#define LVAL   8192
#define KVAL   8192
#define DVAL   256
#define BVAL   8
#define HOFF   (6 * NFFT) // start of H spectra in workspace (floats)

// ---- WMMA f32 16x16x4: D = A(16x4) * B(4x16) + C --------------------------
__device__ __forceinline__ v8f wmma4(v2f a, v2f b, v8f c) {
  // 8 args: (neg_a, A, neg_b, B, c_mod, C, reuse_a, reuse_b); f32 has CNeg
  // only, so neg bits stay 0 and sign handling is done in VALU on fragments.
  return __builtin_amdgcn_wmma_f32_16x16x4_f32(false, a, false, b,
                                               (short)0, c, false, false);
}

// A-operand fragment (16x4 tile, M rows x K cols) per ISA VGPR layout:
//   lane L holds row m0+(L&15); VGPR v + lane-half g -> K = k0 + 2g + v.
__device__ __forceinline__ v2f ldA(const float* p, int stride, int m0, int k0,
                                   int lane) {
  int row = m0 + (lane & 15);
  int col = k0 + ((lane >> 4) << 1);
  const float* q = p + row * stride + col;
  v2f r;
  r.x = q[0];
  r.y = q[1];
  return r;
}

// B-operand fragment (4x16 tile, K rows x N cols): VGPR v + lane-half g holds
// row k0 + 2g + v, column n0+(L&15).
__device__ __forceinline__ v2f ldB(const float* p, int stride, int k0, int n0,
                                   int lane) {
  int g = (lane >> 4) << 1;
  int col = n0 + (lane & 15);
  v2f r;
  r.x = p[(k0 + g + 0) * stride + col];
  r.y = p[(k0 + g + 1) * stride + col];
  return r;
}

// C/D 16x16 tile: VGPR r holds row m0 + r + 8*(L>>4), column n0+(L&15).
__device__ __forceinline__ void stC(float* p, int stride, int m0, int n0,
                                    int lane, v8f acc) {
  int col = n0 + (lane & 15);
  int rbase = m0 + ((lane >> 4) << 3);
#pragma unroll
  for (int r = 0; r < 8; ++r) p[(rbase + r) * stride + col] = acc[r];
}

// ---- 128x128x128 complex GEMM:  OUT = X * Y -------------------------------
// CONST_LEFT:  X = K (DFT matrix, global, stride 128),  Y = Q (LDS, PITCH)
// !CONST_LEFT: X = Q (LDS),                             Y = K (global)
// conj: negate the imaginary part of the constant operand (inverse DFT).
// Re += Xr*Yr + (-Xi)*Yi ; Im += Xr*Yi + Xi*Yr  (4 WMMAs / tile / k-step).
template <bool CONST_LEFT>
__device__ void cgemm128(const float* __restrict__ Kr,
                         const float* __restrict__ Ki, bool conj,
                         const float* Qr, const float* Qi, float* Or,
                         float* Oi) {
  const int lane = threadIdx.x & 31;
  const int wave = threadIdx.x >> 5;
  const int m0 = (wave & 7) << 4;   // 8 m-tiles
  const int nb = (wave >> 3) << 6;  // 2 n-blocks of 4 tiles
  v8f accR[4], accI[4];
#pragma unroll
  for (int j = 0; j < 4; ++j) {
    accR[j] = (v8f){0.f, 0.f, 0.f, 0.f, 0.f, 0.f, 0.f, 0.f};
    accI[j] = (v8f){0.f, 0.f, 0.f, 0.f, 0.f, 0.f, 0.f, 0.f};
  }
  for (int k0 = 0; k0 < NROW; k0 += 4) {
    v2f lr, li;
    if (CONST_LEFT) {
      lr = ldA(Kr, NROW, m0, k0, lane);
      li = ldA(Ki, NROW, m0, k0, lane);
      if (conj) { li.x = -li.x; li.y = -li.y; }
    } else {
      lr = ldA(Qr, PITCH, m0, k0, lane);
      li = ldA(Qi, PITCH, m0, k0, lane);
    }
    v2f liN;
    liN.x = -li.x;
    liN.y = -li.y;
#pragma unroll
    for (int j = 0; j < 4; ++j) {
      int n0 = nb + (j << 4);
      v2f rr, ri;
      if (CONST_LEFT) {
        rr = ldB(Qr, PITCH, k0, n0, lane);
        ri = ldB(Qi, PITCH, k0, n0, lane);
      } else {
        rr = ldB(Kr, NROW, k0, n0, lane);
        ri = ldB(Ki, NROW, k0, n0, lane);
        if (conj) { ri.x = -ri.x; ri.y = -ri.y; }
      }
      accR[j] = wmma4(lr, rr, accR[j]);
      accR[j] = wmma4(liN, ri, accR[j]);
      accI[j] = wmma4(lr, ri, accI[j]);
      accI[j] = wmma4(li, rr, accI[j]);
    }
  }
#pragma unroll
  for (int j = 0; j < 4; ++j) {
    int n0 = nb + (j << 4);
    stC(Or, PITCH, m0, n0, lane, accR[j]);
    stC(Oi, PITCH, m0, n0, lane, accI[j]);
  }
}

// ---- Kernel 0: constants (DFT matrix + twiddles) --------------------------
__global__ void fftconv_init(float* __restrict__ ws) {
  int t = blockIdx.x * blockDim.x + threadIdx.x;
  if (t >= NFFT) return;
  int r = t >> 7, c = t & 127;
  const float twopi = 6.283185307179586476925f;
  int pf = (r * c) & 127;
  float af = twopi * (float)pf * (1.0f / 128.0f);
  ws[0 * NFFT + t] = cosf(af);   // Fr
  ws[1 * NFFT + t] = -sinf(af);  // Fi
  int pt = (r * c) & (NFFT - 1);
  float at = twopi * (float)pt * (1.0f / (float)NFFT);
  float ct = cosf(at), st = sinf(at);
  ws[2 * NFFT + t] = ct;   // Tr  = Re W_N^{rc}
  ws[3 * NFFT + t] = -st;  // Ti
  const float inv = 1.0f / (float)NFFT;
  ws[4 * NFFT + t] = ct * inv;  // Tcr = Re conj(W)^{rc} / N
  ws[5 * NFFT + t] = st * inv;  // Tci
}

// ---- Kernel 1: filter spectra H[d] = FFT16384(pad(h[d])) ------------------
__global__ void __launch_bounds__(WGSZ)
fftconv_hfft(const float* __restrict__ h, float* __restrict__ ws) {
  extern __shared__ float smem[];
  float* Ar = smem;
  float* Ai = smem + PLANE;
  float* Br = smem + 2 * PLANE;
  float* Bi = smem + 3 * PLANE;
  const float* Fr = ws;
  const float* Fi = ws + NFFT;
  const float* Tr = ws + 2 * NFFT;
  const float* Ti = ws + 3 * NFFT;
  const int d = blockIdx.x;
  const int tid = threadIdx.x;

  for (int e = tid; e < NFFT; e += WGSZ) {
    int o = (e >> 7) * PITCH + (e & 127);
    Ar[o] = (e < KVAL) ? h[(size_t)d * KVAL + e] : 0.0f;
    Ai[o] = 0.0f;
  }
  __syncthreads();
  cgemm128<true>(Fr, Fi, false, Ar, Ai, Br, Bi);  // C = F * x
  __syncthreads();
  for (int e = tid; e < NFFT; e += WGSZ) {  // C *= T
    int o = (e >> 7) * PITCH + (e & 127);
    float xr = Br[o], xi = Bi[o], tr = Tr[e], ti = Ti[e];
    Br[o] = xr * tr - xi * ti;
    Bi[o] = xr * ti + xi * tr;
  }
  __syncthreads();
  cgemm128<false>(Fr, Fi, false, Br, Bi, Ar, Ai);  // X = C * F
  __syncthreads();
  float* Hr = ws + HOFF + (size_t)d * (2 * NFFT);
  float* Hi = Hr + NFFT;
  for (int e = tid; e < NFFT; e += WGSZ) {
    int o = (e >> 7) * PITCH + (e & 127);
    Hr[e] = Ar[o];
    Hi[e] = Ai[o];
  }
}

// ---- Kernel 2: per (b,d) row — fwd FFT, multiply, inverse FFT, crop -------
__global__ void __launch_bounds__(WGSZ)
fftconv_main(const float* __restrict__ u, float* __restrict__ out,
             const float* __restrict__ ws) {
  extern __shared__ float smem[];
  float* Ar = smem;
  float* Ai = smem + PLANE;
  float* Br = smem + 2 * PLANE;
  float* Bi = smem + 3 * PLANE;
  const float* Fr = ws;
  const float* Fi = ws + NFFT;
  const float* Tr = ws + 2 * NFFT;
  const float* Ti = ws + 3 * NFFT;
  const float* Tcr = ws + 4 * NFFT;
  const float* Tci = ws + 5 * NFFT;
  const int bid = blockIdx.x;
  const int d = bid & (DVAL - 1);
  const int b = bid >> 8;
  const int tid = threadIdx.x;

  // Load signal row u[b, :, d] (stride D in memory), zero-pad to N.
  for (int e = tid; e < NFFT; e += WGSZ) {
    int o = (e >> 7) * PITCH + (e & 127);
    Ar[o] = (e < LVAL) ? u[((size_t)(b * LVAL + e)) * DVAL + d] : 0.0f;
    Ai[o] = 0.0f;
  }
  __syncthreads();

  // Forward FFT: C = F*x ; C *= T ; X = C*F
  cgemm128<true>(Fr, Fi, false, Ar, Ai, Br, Bi);
  __syncthreads();
  for (int e = tid; e < NFFT; e += WGSZ) {
    int o = (e >> 7) * PITCH + (e & 127);
    float xr = Br[o], xi = Bi[o], tr = Tr[e], ti = Ti[e];
    Br[o] = xr * tr - xi * ti;
    Bi[o] = xr * ti + xi * tr;
  }
  __syncthreads();
  cgemm128<false>(Fr, Fi, false, Br, Bi, Ar, Ai);
  __syncthreads();

  // Pointwise multiply by filter spectrum; NOTE reference row_filter:
  // filter index = (b*D + d) // B = 32*b + d/8.
  const int fd = (b * DVAL + d) >> 3;
  const float* Hr = ws + HOFF + (size_t)fd * (2 * NFFT);
  const float* Hi = Hr + NFFT;
  for (int e = tid; e < NFFT; e += WGSZ) {
    int o = (e >> 7) * PITCH + (e & 127);
    float xr = Ar[o], xi = Ai[o], hr = Hr[e], hi = Hi[e];
    Ar[o] = xr * hr - xi * hi;
    Ai[o] = xr * hi + xi * hr;
  }
  __syncthreads();

  // Inverse FFT: E = Y*conj(F) ; E *= conj(T)/N ; y = conj(F)*E
  cgemm128<false>(Fr, Fi, true, Ar, Ai, Br, Bi);
  __syncthreads();
  for (int e = tid; e < NFFT; e += WGSZ) {
    int o = (e >> 7) * PITCH + (e & 127);
    float xr = Br[o], xi = Bi[o], tr = Tcr[e], ti = Tci[e];
    Br[o] = xr * tr - xi * ti;
    Bi[o] = xr * ti + xi * tr;
  }
  __syncthreads();
  cgemm128<true>(Fr, Fi, true, Br, Bi, Ar, Ai);
  __syncthreads();

  // Crop: y[l] = y_full[l + 4095], real part; out[b, l, d].
  const int start = (KVAL - 1) / 2;  // 4095
  for (int l = tid; l < LVAL; l += WGSZ) {
    int n = l + start;
    int o = (n >> 7) * PITCH + (n & 127);
    out[((size_t)(b * LVAL + l)) * DVAL + d] = Ar[o];
  }
}

// ---------------------------------------------------------------------------
extern "C" void kernel_launch(void* const* d_in, const int* in_sizes, int n_in,
                              void* d_out, int out_size, void* d_ws,
                              size_t ws_size, hipStream_t stream) {
  (void)in_sizes;
  (void)n_in;
  (void)out_size;
  (void)ws_size;  // requires ~34 MB of workspace
  const float* u = (const float*)d_in[0];
  const float* h = (const float*)d_in[1];
  float* out = (float*)d_out;
  float* ws = (float*)d_ws;

  const size_t shbytes = (size_t)4 * PLANE * sizeof(float);  // ~260 KB LDS
  // Opt in to >64KB dynamic LDS (CDNA5: 320 KB per WGP). Deterministic,
  // capture-safe (host-side attribute, not a stream op).
  (void)hipFuncSetAttribute((const void*)fftconv_hfft,
                            hipFuncAttributeMaxDynamicSharedMemorySize,
                            (int)shbytes);
  (void)hipFuncSetAttribute((const void*)fftconv_main,
                            hipFuncAttributeMaxDynamicSharedMemorySize,
                            (int)shbytes);

  fftconv_init<<<NFFT / 256, 256, 0, stream>>>(ws);
  fftconv_hfft<<<DVAL, WGSZ, shbytes, stream>>>(h, ws);
  fftconv_main<<<BVAL * DVAL, WGSZ, shbytes, stream>>>(u, out, ws);
}